// AttentiveKernelMachineLayer_36532991820199
// MI455X (gfx1250) — compile-verified
//
#include <hip/hip_runtime.h>
#include <hip/hip_bf16.h>

typedef __attribute__((ext_vector_type(2))) float v2f;
typedef __attribute__((ext_vector_type(8))) float v8f;

#define WMMA_F32X4(a, b, c) \
    __builtin_amdgcn_wmma_f32_16x16x4_f32(false, (a), false, (b), (short)0, (c), false, false)

// dims
#define NN    2048
#define NIN   512
#define NZ    1024
#define NOUT  64
#define NKEY  20

// ---------------------------------------------------------------------------
// Kernel 0: x2h[n] = 0.5*||x_n||^2, z2h[z] = 0.5*||Z_z||^2.
// One wave (32 lanes) per row; float4 loads; shfl_xor reduction.
// ---------------------------------------------------------------------------
__global__ void rowsq_kernel(const float* __restrict__ x, const float* __restrict__ Z,
                             float* __restrict__ x2h, float* __restrict__ z2h) {
    int wave = threadIdx.x >> 5;
    int lane = threadIdx.x & 31;
    int row  = blockIdx.x * 8 + wave;               // 0..3071
    const float4* src;
    float* dst;
    if (row < NN) {
        src = (const float4*)(x + (size_t)row * NIN);
        dst = x2h + row;
    } else {
        src = (const float4*)(Z + (size_t)(row - NN) * NIN);
        dst = z2h + (row - NN);
    }
    float s = 0.0f;
    for (int i = lane; i < NIN / 4; i += 32) {      // 4 float4 per lane
        float4 v = src[i];
        s += v.x * v.x + v.y * v.y + v.z * v.z + v.w * v.w;
    }
    for (int m = 1; m <= 16; m <<= 1) s += __shfl_xor(s, m, 32);
    if (lane == 0) *dst = 0.5f * s;
}

// ---------------------------------------------------------------------------
// Kernel 1: kxz[n][z] = exp( x_n . Z_z - x2h[n] - z2h[z] )  via WMMA f32.
// One 16x16 tile per wave; K loop over 512 in steps of 4.
// grid(128, 8), block 256.
// ---------------------------------------------------------------------------
__global__ void kxz_kernel(const float* __restrict__ x, const float* __restrict__ Z,
                           const float* __restrict__ x2h, const float* __restrict__ z2h,
                           float* __restrict__ kxz) {
    int wave = threadIdx.x >> 5;
    int lane = threadIdx.x & 31;
    int m    = lane & 15;
    int off  = (lane >> 4) << 1;                    // 0 (lanes 0-15) or 2 (lanes 16-31)
    int hi8  = (lane >> 4) * 8;
    int n0   = blockIdx.x * 16;
    int z0   = (blockIdx.y * 8 + wave) * 16;

    const float* xrow = x + (size_t)(n0 + m) * NIN; // A: row = M index (lane%16)
    const float* zrow = Z + (size_t)(z0 + m) * NIN; // B[i][z] = Z[z][i]: col = lane%16

    v8f c = {};
    for (int k0 = 0; k0 < NIN; k0 += 4) {
        v2f a = *(const v2f*)(xrow + k0 + off);     // K = k0+off, k0+off+1
        v2f b = *(const v2f*)(zrow + k0 + off);
        c = WMMA_F32X4(a, b, c);
    }
    float zh = z2h[z0 + m];
#pragma unroll
    for (int r = 0; r < 8; r++) {
        int row = n0 + r + hi8;
        kxz[(size_t)row * NZ + z0 + m] = __expf(c[r] - x2h[row] - zh);
    }
}

// ---------------------------------------------------------------------------
// Kernel 2: Qt[o][n][key] = sum_i x[n][i] * W_key[o][key][i]   (key 0..19)
// Two 16-col WMMA tiles per wave (keys 0-15, 16-31 with >=20 zero-padded).
// grid(16, 64), block 256 (8 waves -> 8 n-tiles).
// ---------------------------------------------------------------------------
__global__ void qt_kernel(const float* __restrict__ x, const float* __restrict__ W,
                          float* __restrict__ Qt) {
    int o    = blockIdx.y;
    int wave = threadIdx.x >> 5;
    int lane = threadIdx.x & 31;
    int m    = lane & 15;
    int off  = (lane >> 4) << 1;
    int hi8  = (lane >> 4) * 8;
    int n0   = (blockIdx.x * 8 + wave) * 16;

    const float* xrow = x + (size_t)(n0 + m) * NIN;
    const float* Wo   = W + (size_t)o * NKEY * NIN;

    v8f c0 = {}, c1 = {};
    for (int k0 = 0; k0 < NIN; k0 += 4) {
        v2f a  = *(const v2f*)(xrow + k0 + off);
        v2f b0 = *(const v2f*)(Wo + (size_t)m * NIN + k0 + off);          // keys 0-15
        v2f b1 = {0.0f, 0.0f};
        if (m < NKEY - 16)                                                // keys 16-19
            b1 = *(const v2f*)(Wo + (size_t)(16 + m) * NIN + k0 + off);
        c0 = WMMA_F32X4(a, b0, c0);
        c1 = WMMA_F32X4(a, b1, c1);
    }
    float* qbase = Qt + ((size_t)o * NN + n0) * NKEY;
#pragma unroll
    for (int r = 0; r < 8; r++) {
        int row = r + hi8;
        qbase[(size_t)row * NKEY + m] = c0[r];
        if (m < NKEY - 16) qbase[(size_t)row * NKEY + 16 + m] = c1[r];
    }
}

// ---------------------------------------------------------------------------
// Kernel 3: Km[o][key][z] = W_key[o][key][:] . Z[z][:]
// One thread per output (1.31M); operands are L2-resident (2.6 MB + 2 MB).
// ---------------------------------------------------------------------------
__global__ void kproj_kernel(const float* __restrict__ W, const float* __restrict__ Z,
                             float* __restrict__ Km) {
    int idx = blockIdx.x * 256 + threadIdx.x;
    if (idx >= NOUT * NKEY * NZ) return;
    int z  = idx & (NZ - 1);
    int ok = idx >> 10;                              // o*20 + key
    const float4* wr = (const float4*)(W + (size_t)ok * NIN);
    const float4* zr = (const float4*)(Z + (size_t)z * NIN);
    float s = 0.0f;
    for (int i = 0; i < NIN / 4; i++) {
        float4 a = wr[i], b = zr[i];
        s += a.x * b.x + a.y * b.y + a.z * b.z + a.w * b.w;
    }
    Km[idx] = s;
}

// ---------------------------------------------------------------------------
// Kernel 4 (fused): per block (o, 16-row n-tile):
//   V = Qt_tile @ Km[o]  (WMMA, K=20), + gumbel (single HBM pass),
//   softmax over z=1024 (cross-wave via shfl + LDS),
//   out[n][o] = sum_z kxz*e / sum_z e.
// grid(128, 64), block 256 (8 waves x 128 z-cols each).
// ---------------------------------------------------------------------------
__global__ void fused_kernel(const float* __restrict__ Qt, const float* __restrict__ Km,
                             const float* __restrict__ kxz, const float* __restrict__ g,
                             float* __restrict__ out) {
    __shared__ float sred[16][8];
    __shared__ float srow[16];
    __shared__ float ssum[16][8][2];

    int o    = blockIdx.y;
    int n0   = blockIdx.x * 16;
    int wave = threadIdx.x >> 5;
    int lane = threadIdx.x & 31;
    int m    = lane & 15;
    int off  = (lane >> 4) << 1;
    int hi8  = (lane >> 4) * 8;

    // A fragments: Qt rows n0..n0+15, K = 20 (5 steps of 4)
    const float* qrow = Qt + ((size_t)o * NN + n0 + m) * NKEY;
    v2f a[5];
#pragma unroll
    for (int kk = 0; kk < 5; kk++) a[kk] = *(const v2f*)(qrow + kk * 4 + off);

    const float* Ko   = Km + (size_t)o * NKEY * NZ;
    const float* grow = g + ((size_t)o * NN + n0) * NZ;

    v8f acc[8];
#pragma unroll
    for (int zt = 0; zt < 8; zt++) {
        int z = (wave * 8 + zt) * 16 + m;
        v8f c = {};
#pragma unroll
        for (int kk = 0; kk < 5; kk++) {
            int k0 = kk * 4 + off;
            v2f b = { Ko[(size_t)k0 * NZ + z], Ko[(size_t)(k0 + 1) * NZ + z] };
            c = WMMA_F32X4(a[kk], b, c);
        }
#pragma unroll
        for (int r = 0; r < 8; r++)                 // V + gumbel (only 512MB stream)
            c[r] += grow[(size_t)(r + hi8) * NZ + z];
        acc[zt] = c;
    }

    // ---- row max over z (wave-local shfl within 16-lane half, then LDS) ----
#pragma unroll
    for (int r = 0; r < 8; r++) {
        float mx = acc[0][r];
#pragma unroll
        for (int zt = 1; zt < 8; zt++) mx = fmaxf(mx, acc[zt][r]);
        for (int s = 1; s <= 8; s <<= 1) mx = fmaxf(mx, __shfl_xor(mx, s, 32));
        if (m == 0) sred[r + hi8][wave] = mx;
    }
    __syncthreads();
    if (threadIdx.x < 16) {
        float gmx = sred[threadIdx.x][0];
        for (int w = 1; w < 8; w++) gmx = fmaxf(gmx, sred[threadIdx.x][w]);
        srow[threadIdx.x] = gmx;
    }
    __syncthreads();
    float rmax[8];
#pragma unroll
    for (int r = 0; r < 8; r++) rmax[r] = srow[r + hi8];

    // ---- exp, sum(e), sum(e*kxz) ----
#pragma unroll
    for (int r = 0; r < 8; r++) {
        int row = r + hi8;
        float se = 0.0f, sk = 0.0f;
#pragma unroll
        for (int zt = 0; zt < 8; zt++) {
            int z = (wave * 8 + zt) * 16 + m;
            float e = __expf(acc[zt][r] - rmax[r]);
            se += e;
            sk += e * kxz[(size_t)(n0 + row) * NZ + z];
        }
        for (int s = 1; s <= 8; s <<= 1) {
            se += __shfl_xor(se, s, 32);
            sk += __shfl_xor(sk, s, 32);
        }
        if (m == 0) { ssum[row][wave][0] = se; ssum[row][wave][1] = sk; }
    }
    __syncthreads();
    if (threadIdx.x < 16) {
        float te = 0.0f, tk = 0.0f;
        for (int w = 0; w < 8; w++) { te += ssum[threadIdx.x][w][0]; tk += ssum[threadIdx.x][w][1]; }
        out[(size_t)(n0 + threadIdx.x) * NOUT + o] = tk / te;
    }
}

// ---------------------------------------------------------------------------
extern "C" void kernel_launch(void* const* d_in, const int* in_sizes, int n_in,
                              void* d_out, int out_size, void* d_ws, size_t ws_size,
                              hipStream_t stream) {
    const float* x = (const float*)d_in[0];   // [2048, 512]
    const float* Z = (const float*)d_in[1];   // [1024, 512]
    const float* W = (const float*)d_in[2];   // [64, 20, 512]
    const float* g = (const float*)d_in[3];   // [64, 2048, 1024]
    float* out = (float*)d_out;               // [2048, 64]

    float* kxz = (float*)d_ws;                                  // 2048*1024      (8 MB)
    float* Qt  = kxz + (size_t)NN * NZ;                         // 64*2048*20     (10.5 MB)
    float* Km  = Qt + (size_t)NOUT * NN * NKEY;                 // 64*20*1024     (5.25 MB)
    float* x2h = Km + (size_t)NOUT * NKEY * NZ;                 // 2048
    float* z2h = x2h + NN;                                      // 1024

    rowsq_kernel<<<dim3((NN + NZ) / 8), 256, 0, stream>>>(x, Z, x2h, z2h);
    kxz_kernel<<<dim3(NN / 16, NZ / 128), 256, 0, stream>>>(x, Z, x2h, z2h, kxz);
    qt_kernel<<<dim3(NN / 128, NOUT), 256, 0, stream>>>(x, W, Qt);
    kproj_kernel<<<dim3((NOUT * NKEY * NZ + 255) / 256), 256, 0, stream>>>(W, Z, Km);
    fused_kernel<<<dim3(NN / 16, NOUT), 256, 0, stream>>>(Qt, Km, kxz, g, out);
}